// SetAbstraction_47545287967128
// MI455X (gfx1250) — compile-verified
//
#include <hip/hip_runtime.h>

typedef __attribute__((ext_vector_type(2))) float v2f;
typedef __attribute__((ext_vector_type(8))) float v8f;

// Problem constants (from reference setup_inputs)
#define B_    16
#define N1_   4096
#define N2_   1024
#define K_    32
#define C1_   64
#define MTOT  (B_ * N2_ * K_)       // 524288
#define GTOT  (B_ * N2_)            // 16384 k-groups

// ---------------------------------------------------------------------------
// conv1: fused gather + concat + GEMM (W0[64x67] * X[67xM]) + raw store +
//        per-channel sum / sumsq accumulation for BN.
// Block: 256 threads = 8 waves, 128 points (16 points per wave).
// ---------------------------------------------------------------------------
__global__ __launch_bounds__(256)
void conv1_kernel(const float* __restrict__ in_xyz,    // [16][3][4096]
                  const float* __restrict__ out_xyz,   // [16][3][1024]
                  const float* __restrict__ in_feat,   // [16][64][4096]
                  const int*   __restrict__ nbr,       // flat [M]
                  const float* __restrict__ W0,        // [64][67]
                  float* __restrict__ Y0,              // pairs [32][M][2]
                  float* __restrict__ statSum,         // [64]
                  float* __restrict__ statSq)          // [64]
{
    __shared__ float Xs[34 * 128 * 2];   // pair-interleaved activation tile
    __shared__ float Ws[64 * 68];        // padded weights (col 67 = 0)
    __shared__ float bsum[64], bsq[64];

    const int tid = threadIdx.x;
    const int m0  = blockIdx.x * 128;
    const int b   = m0 >> 15;            // / (N2_*K_)
    const int rem = m0 & 32767;

    if (tid < 64) { bsum[tid] = 0.f; bsq[tid] = 0.f; }
    for (int i = tid; i < 64 * 68; i += 256) {
        int row = i / 68, col = i - row * 68;
        Ws[i] = (col < 67) ? W0[row * 67 + col] : 0.f;
    }

    // gather: 2 threads per point (halves of channel pairs)
    {
        const int p = tid & 127;
        const int h = tid >> 7;
        const int m = m0 + p;
        const int idx = nbr[m];
        const float* fb = in_feat + (size_t)b * 64 * N1_ + idx;
        if (h == 0) {
            #pragma unroll
            for (int pr = 0; pr < 17; ++pr) {
                Xs[(pr * 128 + p) * 2 + 0] = fb[(2 * pr + 0) * N1_];
                Xs[(pr * 128 + p) * 2 + 1] = fb[(2 * pr + 1) * N1_];
            }
        } else {
            #pragma unroll
            for (int pr = 17; pr < 32; ++pr) {
                Xs[(pr * 128 + p) * 2 + 0] = fb[(2 * pr + 0) * N1_];
                Xs[(pr * 128 + p) * 2 + 1] = fb[(2 * pr + 1) * N1_];
            }
            const int n2 = (rem + p) >> 5;
            const float* xb = in_xyz  + (size_t)b * 3 * N1_ + idx;
            const float* ob = out_xyz + (size_t)b * 3 * N2_ + n2;
            Xs[(32 * 128 + p) * 2 + 0] = xb[0]        - ob[0];
            Xs[(32 * 128 + p) * 2 + 1] = xb[N1_]      - ob[N2_];
            Xs[(33 * 128 + p) * 2 + 0] = xb[2 * N1_]  - ob[2 * N2_];
            Xs[(33 * 128 + p) * 2 + 1] = 0.f;
        }
    }
    __syncthreads();

    const int wave = tid >> 5;
    const int lane = tid & 31;
    const int l15  = lane & 15;
    const int hi   = lane >> 4;
    const int col  = wave * 16 + l15;    // local point column

    v8f acc[4];
    #pragma unroll
    for (int r = 0; r < 4; ++r)
        #pragma unroll
        for (int j = 0; j < 8; ++j) acc[r][j] = 0.f;

    for (int kk = 0; kk < 68; kk += 4) {
        // B fragment: rows kk+2*hi, kk+2*hi+1 ; cols = l15
        v2f bfrag = *(const v2f*)&Xs[((kk >> 1) + hi) * 256 + col * 2];
        #pragma unroll
        for (int r = 0; r < 4; ++r) {
            v2f afrag = *(const v2f*)&Ws[(r * 16 + l15) * 68 + kk + 2 * hi];
            acc[r] = __builtin_amdgcn_wmma_f32_16x16x4_f32(
                false, afrag, false, bfrag, (short)0, acc[r], false, false);
        }
    }

    // epilogue: store raw y0 (pair layout) + block BN partial sums
    #pragma unroll
    for (int r = 0; r < 4; ++r) {
        #pragma unroll
        for (int j = 0; j < 8; j += 2) {
            v2f v; v.x = acc[r][j]; v.y = acc[r][j + 1];
            int pairc = r * 8 + (j >> 1) + hi * 4;
            *(v2f*)&Y0[(size_t)pairc * (2 * MTOT) + (size_t)(m0 + col) * 2] = v;
        }
        #pragma unroll
        for (int j = 0; j < 8; ++j) {
            float s = acc[r][j];
            float q = s * s;
            for (int off = 1; off < 16; off <<= 1) {
                s += __shfl_xor(s, off, 32);
                q += __shfl_xor(q, off, 32);
            }
            if (l15 == 0) {
                int c = r * 16 + j + 8 * hi;
                atomicAdd(&bsum[c], s);
                atomicAdd(&bsq[c], q);
            }
        }
    }
    __syncthreads();
    if (tid < 64) {
        atomicAdd(&statSum[tid], bsum[tid]);
        atomicAdd(&statSq[tid],  bsq[tid]);
    }
}

// ---------------------------------------------------------------------------
// conv2: y1 = W1[64x64] * relu(s0*y0 + t0); raw store + BN stats.
// ---------------------------------------------------------------------------
__global__ __launch_bounds__(256)
void conv2_kernel(const float* __restrict__ X,    // Y0 pairs [32][M][2] raw
                  const float* __restrict__ W,    // [64][64]
                  const float* __restrict__ st,   // s[0..63], t[64..127]
                  float* __restrict__ Y,          // Y1 pairs [32][M][2]
                  float* __restrict__ statSum,
                  float* __restrict__ statSq)
{
    __shared__ float Ws[64 * 64];
    __shared__ float ss[64], ts[64];
    __shared__ float bsum[64], bsq[64];

    const int tid = threadIdx.x;
    if (tid < 64) { ss[tid] = st[tid]; ts[tid] = st[64 + tid]; bsum[tid] = 0.f; bsq[tid] = 0.f; }
    for (int i = tid; i < 64 * 64; i += 256) Ws[i] = W[i];
    __syncthreads();

    const int m0   = blockIdx.x * 128;
    const int wave = tid >> 5;
    const int lane = tid & 31;
    const int l15  = lane & 15;
    const int hi   = lane >> 4;
    const int col  = wave * 16 + l15;

    v8f acc[4];
    #pragma unroll
    for (int r = 0; r < 4; ++r)
        #pragma unroll
        for (int j = 0; j < 8; ++j) acc[r][j] = 0.f;

    for (int kk = 0; kk < 64; kk += 4) {
        const int row0 = kk + 2 * hi;
        v2f raw = *(const v2f*)&X[(size_t)((kk >> 1) + hi) * (2 * MTOT) + (size_t)(m0 + col) * 2];
        v2f bfrag;
        bfrag.x = fmaxf(raw.x * ss[row0]     + ts[row0],     0.f);
        bfrag.y = fmaxf(raw.y * ss[row0 + 1] + ts[row0 + 1], 0.f);
        #pragma unroll
        for (int r = 0; r < 4; ++r) {
            v2f afrag = *(const v2f*)&Ws[(r * 16 + l15) * 64 + kk + 2 * hi];
            acc[r] = __builtin_amdgcn_wmma_f32_16x16x4_f32(
                false, afrag, false, bfrag, (short)0, acc[r], false, false);
        }
    }

    #pragma unroll
    for (int r = 0; r < 4; ++r) {
        #pragma unroll
        for (int j = 0; j < 8; j += 2) {
            v2f v; v.x = acc[r][j]; v.y = acc[r][j + 1];
            int pairc = r * 8 + (j >> 1) + hi * 4;
            *(v2f*)&Y[(size_t)pairc * (2 * MTOT) + (size_t)(m0 + col) * 2] = v;
        }
        #pragma unroll
        for (int j = 0; j < 8; ++j) {
            float s = acc[r][j];
            float q = s * s;
            for (int off = 1; off < 16; off <<= 1) {
                s += __shfl_xor(s, off, 32);
                q += __shfl_xor(q, off, 32);
            }
            if (l15 == 0) {
                int c = r * 16 + j + 8 * hi;
                atomicAdd(&bsum[c], s);
                atomicAdd(&bsq[c], q);
            }
        }
    }
    __syncthreads();
    if (tid < 64) {
        atomicAdd(&statSum[tid], bsum[tid]);
        atomicAdd(&statSq[tid],  bsq[tid]);
    }
}

// ---------------------------------------------------------------------------
// conv3: y2 = W2[128x64] * relu(s1*y1 + t1); each wave owns one k-group
// (32 points) and reduces max & min of raw y2 over k, plus BN stats.
// Block: 256 threads = 8 waves = 256 points = 8 k-groups.
// ---------------------------------------------------------------------------
__global__ __launch_bounds__(256)
void conv3_kernel(const float* __restrict__ X,    // Y1 pairs [32][M][2] raw
                  const float* __restrict__ W,    // [128][64]
                  const float* __restrict__ st,   // s1/t1
                  float* __restrict__ mm,         // [G][256]: max[c], min[128+c]
                  float* __restrict__ statSum,    // [128]
                  float* __restrict__ statSq)     // [128]
{
    __shared__ float Ws[128 * 64];
    __shared__ float ss[64], ts[64];
    __shared__ float bsum[128], bsq[128];
    __shared__ float stage[8][256];

    const int tid = threadIdx.x;
    if (tid < 64)  { ss[tid] = st[tid]; ts[tid] = st[64 + tid]; }
    if (tid < 128) { bsum[tid] = 0.f; bsq[tid] = 0.f; }
    for (int i = tid; i < 128 * 64; i += 256) Ws[i] = W[i];
    __syncthreads();

    const int wave  = tid >> 5;
    const int lane  = tid & 31;
    const int l15   = lane & 15;
    const int hi    = lane >> 4;
    const int mbase = blockIdx.x * 256 + wave * 32;   // k-group base
    float* mystage  = stage[wave];

    for (int half = 0; half < 2; ++half) {
        const int gm = mbase + half * 16 + l15;       // global point column

        v8f acc[8];
        #pragma unroll
        for (int r = 0; r < 8; ++r)
            #pragma unroll
            for (int j = 0; j < 8; ++j) acc[r][j] = 0.f;

        for (int kk = 0; kk < 64; kk += 4) {
            const int row0 = kk + 2 * hi;
            v2f raw = *(const v2f*)&X[(size_t)((kk >> 1) + hi) * (2 * MTOT) + (size_t)gm * 2];
            v2f bfrag;
            bfrag.x = fmaxf(raw.x * ss[row0]     + ts[row0],     0.f);
            bfrag.y = fmaxf(raw.y * ss[row0 + 1] + ts[row0 + 1], 0.f);
            #pragma unroll
            for (int r = 0; r < 8; ++r) {
                v2f afrag = *(const v2f*)&Ws[(r * 16 + l15) * 64 + kk + 2 * hi];
                acc[r] = __builtin_amdgcn_wmma_f32_16x16x4_f32(
                    false, afrag, false, bfrag, (short)0, acc[r], false, false);
            }
        }

        #pragma unroll
        for (int r = 0; r < 8; ++r) {
            #pragma unroll
            for (int j = 0; j < 8; ++j) {
                float v  = acc[r][j];
                float s  = v, q = v * v, mx = v, mn = v;
                for (int off = 1; off < 16; off <<= 1) {
                    s  += __shfl_xor(s,  off, 32);
                    q  += __shfl_xor(q,  off, 32);
                    mx  = fmaxf(mx, __shfl_xor(mx, off, 32));
                    mn  = fminf(mn, __shfl_xor(mn, off, 32));
                }
                if (l15 == 0) {
                    int c = r * 16 + j + 8 * hi;
                    atomicAdd(&bsum[c], s);
                    atomicAdd(&bsq[c],  q);
                    if (half == 0) {
                        mystage[c]       = mx;
                        mystage[128 + c] = mn;
                    } else {
                        mystage[c]       = fmaxf(mystage[c],       mx);
                        mystage[128 + c] = fminf(mystage[128 + c], mn);
                    }
                }
            }
        }
        __syncthreads();   // also orders the wave-local LDS stage updates
    }

    // coalesced writeback of per-k-group max/min
    const int g = mbase >> 5;
    for (int i = lane; i < 256; i += 32)
        mm[(size_t)g * 256 + i] = mystage[i];

    if (tid < 128) {
        atomicAdd(&statSum[tid], bsum[tid]);
        atomicAdd(&statSq[tid],  bsq[tid]);
    }
}

// ---------------------------------------------------------------------------
// BN stats -> per-channel affine (s, t)
// ---------------------------------------------------------------------------
__global__ void bn_stats_kernel(const float* __restrict__ sum,
                                const float* __restrict__ sq,
                                const float* __restrict__ gamma,
                                const float* __restrict__ beta,
                                float* __restrict__ st, int C, float invM)
{
    int c = threadIdx.x;
    if (c < C) {
        float mean = sum[c] * invM;
        float var  = sq[c] * invM - mean * mean;
        float inv  = rsqrtf(var + 1e-5f);
        float s    = gamma[c] * inv;
        float t    = beta[c] - mean * s;
        st[c]     = s;
        st[C + c] = t;
    }
}

// ---------------------------------------------------------------------------
// final: out[b][c][n2] = relu(s2 * (s2>=0 ? max : min) + t2)
// ---------------------------------------------------------------------------
__global__ __launch_bounds__(128)
void final_kernel(const float* __restrict__ mm,
                  const float* __restrict__ st2,
                  float* __restrict__ out)
{
    const int g = blockIdx.x;        // 0..16383
    const int c = threadIdx.x;       // 0..127
    float s  = st2[c];
    float t  = st2[128 + c];
    float mx = mm[(size_t)g * 256 + c];
    float mn = mm[(size_t)g * 256 + 128 + c];
    float v  = (s >= 0.f) ? mx : mn;
    float y  = fmaxf(v * s + t, 0.f);
    const int b = g >> 10, n2 = g & 1023;
    out[((size_t)b * 128 + c) * N2_ + n2] = y;
}

// ---------------------------------------------------------------------------
extern "C" void kernel_launch(void* const* d_in, const int* in_sizes, int n_in,
                              void* d_out, int out_size, void* d_ws, size_t ws_size,
                              hipStream_t stream)
{
    const float* in_xyz  = (const float*)d_in[0];
    const float* out_xyz = (const float*)d_in[1];
    const float* in_feat = (const float*)d_in[2];
    const int*   nbr     = (const int*)  d_in[3];
    const float* W0 = (const float*)d_in[4];
    const float* g0 = (const float*)d_in[5];
    const float* b0 = (const float*)d_in[6];
    const float* W1 = (const float*)d_in[7];
    const float* g1 = (const float*)d_in[8];
    const float* b1 = (const float*)d_in[9];
    const float* W2 = (const float*)d_in[10];
    const float* g2 = (const float*)d_in[11];
    const float* b2 = (const float*)d_in[12];

    float* ws      = (float*)d_ws;
    float* statRaw = ws;                                    // 512 floats
    float* st0     = ws + 512;                              // 128
    float* st1     = ws + 640;                              // 128
    float* st2     = ws + 768;                              // 256
    float* mm      = ws + 1024;                             // 16384*256
    float* Y0      = mm + (size_t)GTOT * 256;               // 64*M
    float* Y1      = Y0 + (size_t)64 * MTOT;                // 64*M

    const float invM = 1.0f / (float)MTOT;

    hipMemsetAsync(statRaw, 0, 512 * sizeof(float), stream);

    conv1_kernel<<<MTOT / 128, 256, 0, stream>>>(in_xyz, out_xyz, in_feat, nbr,
                                                 W0, Y0, statRaw + 0, statRaw + 64);
    bn_stats_kernel<<<1, 64, 0, stream>>>(statRaw + 0, statRaw + 64, g0, b0, st0, 64, invM);

    conv2_kernel<<<MTOT / 128, 256, 0, stream>>>(Y0, W1, st0, Y1,
                                                 statRaw + 128, statRaw + 192);
    bn_stats_kernel<<<1, 64, 0, stream>>>(statRaw + 128, statRaw + 192, g1, b1, st1, 64, invM);

    conv3_kernel<<<MTOT / 256, 256, 0, stream>>>(Y1, W2, st1, mm,
                                                 statRaw + 256, statRaw + 384);
    bn_stats_kernel<<<1, 128, 0, stream>>>(statRaw + 256, statRaw + 384, g2, b2, st2, 128, invM);

    final_kernel<<<GTOT, 128, 0, stream>>>(mm, st2, (float*)d_out);
}